// GCN_37546604102454
// MI455X (gfx1250) — compile-verified
//
#include <hip/hip_runtime.h>
#include <hip/hip_bf16.h>

typedef __attribute__((ext_vector_type(2))) float v2f;
typedef __attribute__((ext_vector_type(8))) float v8f;

// ---------------------------------------------------------------------------
// Degree / normalization kernels
// ---------------------------------------------------------------------------
__global__ __launch_bounds__(256) void set_val_kernel(float* __restrict__ p,
                                                      float v, int n) {
  int i = blockIdx.x * blockDim.x + threadIdx.x;
  if (i < n) p[i] = v;
}

__global__ __launch_bounds__(256) void deg_edges_kernel(
    const long long* __restrict__ dst, float* __restrict__ deg, int nedges) {
  int e = blockIdx.x * blockDim.x + threadIdx.x;
  if (e < nedges) atomicAdd(deg + dst[e], 1.0f);
}

__global__ __launch_bounds__(256) void rsqrt_kernel(const float* __restrict__ deg,
                                                    float* __restrict__ dinv,
                                                    int n) {
  int i = blockIdx.x * blockDim.x + threadIdx.x;
  if (i < n) dinv[i] = rsqrtf(deg[i]);  // deg >= 1 (self loop), no zero guard
}

// ---------------------------------------------------------------------------
// WMMA GEMM: C[nrows,NC] = A[nrows,K] @ W[K,NC] (+ bias), full fp32 via
// V_WMMA_F32_16X16X4_F32. Block = 256 threads = 8 waves; wave owns a 16-row
// strip and all NC/16 column tiles.
//
// W is staged into LDS K-pair interleaved: sW[(k/2)*NC + col] holds
// {W[k][col], W[k+1][col]} as a float2, so a B fragment (v2f: rows kb,kb+1 at
// one column) is a single aligned ds_load_b64 into an even VGPR pair — no
// shuffle movs in the inner loop.
// ---------------------------------------------------------------------------
template <int K, int NC>
__global__ __launch_bounds__(256) void gemm_wmma_f32(
    const float* __restrict__ A, const float* __restrict__ W,
    const float* __restrict__ bias, float* __restrict__ C, int nrows) {
  constexpr int KC = (K < 128) ? K : 128;  // LDS chunk depth (<=64KB)
  constexpr int NT = NC / 16;              // column tiles per wave
  constexpr int PAIRS = (KC / 2) * NC;     // float2 entries per chunk
  __shared__ __align__(16) v2f sW[PAIRS];

  const int tid    = threadIdx.x;
  const int wave   = tid >> 5;
  const int lane   = tid & 31;
  const int laneLo = lane & 15;
  const int laneHi = lane >> 4;  // 0: K={k,k+1}/M=0..7, 1: K={k+2,k+3}/M=8..15

  const int nTiles = (nrows + 15) >> 4;
  const int tile   = blockIdx.x * 8 + wave;
  const bool live  = tile < nTiles;
  const int t      = live ? tile : (nTiles - 1);  // clamp: keep EXEC full
  const int row0   = t << 4;

  v8f acc[NT];
#pragma unroll
  for (int i = 0; i < NT; ++i)
    acc[i] = (v8f){0.f, 0.f, 0.f, 0.f, 0.f, 0.f, 0.f, 0.f};

  // A fragment base: lanes 0-15 hold rows row0..row0+15 at K={k,k+1},
  // lanes 16-31 same rows at K={k+2,k+3}  (ISA 32-bit A 16x4 layout).
  const float* arow = A + (size_t)(row0 + laneLo) * K + laneHi * 2;

  // per-lane LDS read base (in float2 units): row (k/2 + laneHi), col laneLo
  const v2f* sread = sW + (size_t)laneHi * NC + laneLo;

  for (int kc = 0; kc < K; kc += KC) {
    __syncthreads();
    // stage W[kc..kc+KC) x NC into LDS, K-pair interleaved.
    // thread p: kp = p/NC (k-pair row), col = p%NC; two coalesced row reads,
    // one ds_store_b64; consecutive threads -> consecutive 8B LDS addrs.
    for (int p = tid; p < PAIRS; p += 256) {
      const int kp  = p / NC;
      const int col = p - kp * NC;
      v2f w2;
      w2.x = W[(size_t)(kc + 2 * kp) * NC + col];
      w2.y = W[(size_t)(kc + 2 * kp + 1) * NC + col];
      sW[p] = w2;
    }
    __syncthreads();

#pragma unroll 4
    for (int k = 0; k < KC; k += 4) {
      v2f a = *(const v2f*)(arow + kc + k);
      // gather all B fragments first so ds_loads batch under one dscnt wait
      v2f bf[NT];
#pragma unroll
      for (int ct = 0; ct < NT; ++ct)
        bf[ct] = sread[(k >> 1) * NC + ct * 16];
#pragma unroll
      for (int ct = 0; ct < NT; ++ct)
        acc[ct] = __builtin_amdgcn_wmma_f32_16x16x4_f32(
            false, a, false, bf[ct], (short)0, acc[ct], false, false);
    }
  }

  if (live) {
#pragma unroll
    for (int ct = 0; ct < NT; ++ct) {
      const int c = ct * 16 + laneLo;
      const float bv = bias ? bias[c] : 0.f;
#pragma unroll
      for (int v = 0; v < 8; ++v) {
        const int r = row0 + laneHi * 8 + v;  // C/D layout: VGPR v -> M
        if (r < nrows) C[(size_t)r * NC + c] = acc[ct][v] + bv;
      }
    }
  }
}

// ---------------------------------------------------------------------------
// Aggregation: agg = bias + xw*dinv^2 (self loop), then wave-per-edge
// scatter-add of xw[src]*dinv[src]*dinv[dst] into agg[dst].
// ---------------------------------------------------------------------------
__global__ __launch_bounds__(256) void agg_init_kernel(
    const float* __restrict__ xw, const float* __restrict__ bias,
    const float* __restrict__ dinv, float* __restrict__ agg, int total) {
  int i = blockIdx.x * blockDim.x + threadIdx.x;
  if (i < total) {
    int r = i >> 7;  // NC = 128
    int c = i & 127;
    float di = dinv[r];
    agg[i] = bias[c] + xw[i] * di * di;
  }
}

__global__ __launch_bounds__(256) void agg_edges_kernel(
    const float* __restrict__ xw, const long long* __restrict__ src,
    const long long* __restrict__ dst, const float* __restrict__ dinv,
    float* __restrict__ agg, int nedges) {
  int gw = (blockIdx.x * blockDim.x + threadIdx.x) >> 5;  // wave per edge
  int lane = threadIdx.x & 31;
  if (gw >= nedges) return;
  long long s = src[gw];
  long long d = dst[gw];
  float w = dinv[s] * dinv[d];
  float4 m = ((const float4*)(xw + (size_t)s * 128))[lane];  // 512B/wave, L2 hit
  float* a = agg + (size_t)d * 128 + lane * 4;
  atomicAdd(a + 0, m.x * w);
  atomicAdd(a + 1, m.y * w);
  atomicAdd(a + 2, m.z * w);
  atomicAdd(a + 3, m.w * w);
}

__global__ __launch_bounds__(256) void relu_kernel(const float* __restrict__ in,
                                                   float* __restrict__ out,
                                                   int total) {
  int i = blockIdx.x * blockDim.x + threadIdx.x;
  if (i < total) out[i] = fmaxf(in[i], 0.f);
}

// ---------------------------------------------------------------------------
// Launch
// ---------------------------------------------------------------------------
extern "C" void kernel_launch(void* const* d_in, const int* in_sizes, int n_in,
                              void* d_out, int out_size, void* d_ws,
                              size_t ws_size, hipStream_t stream) {
  const float*     fts = (const float*)d_in[0];
  const long long* ei  = (const long long*)d_in[1];
  const float*     W1  = (const float*)d_in[2];
  const float*     b1  = (const float*)d_in[3];
  const float*     W2  = (const float*)d_in[4];
  const float*     b2  = (const float*)d_in[5];
  const float*     Wc  = (const float*)d_in[6];
  const float*     bc  = (const float*)d_in[7];

  const int N = in_sizes[0] / 256;  // IN_DIM = 256
  const int E = in_sizes[1] / 2;    // edge_index is (2, E)
  const long long* src = ei;
  const long long* dst = ei + E;

  float* out_final = (float*)d_out;               // [N,16]
  float* x_out     = out_final + (size_t)N * 16;  // [N,128] second output

  // workspace layout
  float* deg  = (float*)d_ws;
  float* dinv = deg + N;
  float* xw   = dinv + N;               // [N,128]
  float* agg  = xw + (size_t)N * 128;   // [N,128]
  float* h1   = agg + (size_t)N * 128;  // [N,128]

  const int nTiles = (N + 15) >> 4;
  const int gBlks  = (nTiles + 7) / 8;
  const int nVec   = N * 128;
  const int vBlks  = (nVec + 255) / 256;
  const int nBlks  = (N + 255) / 256;
  const int eBlks  = (E + 255) / 256;
  const int ewBlks = (int)(((size_t)E * 32 + 255) / 256);  // wave per edge

  // degree + symmetric normalization
  set_val_kernel<<<nBlks, 256, 0, stream>>>(deg, 1.0f, N);
  deg_edges_kernel<<<eBlks, 256, 0, stream>>>(dst, deg, E);
  rsqrt_kernel<<<nBlks, 256, 0, stream>>>(deg, dinv, N);

  // layer 1: fts @ W1 -> aggregate -> relu
  gemm_wmma_f32<256, 128><<<gBlks, 256, 0, stream>>>(fts, W1, nullptr, xw, N);
  agg_init_kernel<<<vBlks, 256, 0, stream>>>(xw, b1, dinv, agg, nVec);
  agg_edges_kernel<<<ewBlks, 256, 0, stream>>>(xw, src, dst, dinv, agg, E);
  relu_kernel<<<vBlks, 256, 0, stream>>>(agg, h1, nVec);

  // layer 2: h1 @ W2 -> aggregate -> relu (x output lands in d_out)
  gemm_wmma_f32<128, 128><<<gBlks, 256, 0, stream>>>(h1, W2, nullptr, xw, N);
  agg_init_kernel<<<vBlks, 256, 0, stream>>>(xw, b2, dinv, agg, nVec);
  agg_edges_kernel<<<ewBlks, 256, 0, stream>>>(xw, src, dst, dinv, agg, E);
  relu_kernel<<<vBlks, 256, 0, stream>>>(agg, x_out, nVec);

  // head: x @ Wc + bc
  gemm_wmma_f32<128, 16><<<gBlks, 256, 0, stream>>>(x_out, Wc, bc, out_final, N);
}